// CRF_89077621719429
// MI455X (gfx1250) — compile-verified
//
#include <hip/hip_runtime.h>
#include <math.h>

typedef float v2f __attribute__((ext_vector_type(2)));
typedef float v8f __attribute__((ext_vector_type(8)));

#define NEGV   (-10000.0f)
#define NTAG   6
#define STARTS 4
#define STOPS  5
#define BSZ    512
#define TLEN   2048
#define CHUNK  64
#define NCH    (TLEN / CHUNK)   /* 32 chunks */

// ds_swizzle with compile-time pattern (group-of-32: and=0x1f, or=0, xor=X)
template <int OFF>
__device__ __forceinline__ float swz(float x) {
  return __int_as_float(__builtin_amdgcn_ds_swizzle(__float_as_int(x), OFF));
}
#define SWZ_X16 0x401F
#define SWZ_X8  0x201F
#define SWZ_X4  0x101F
#define SWZ_X2  0x081F
#define SWZ_X1  0x041F

// robust fast tanh: 1 - 2/(e^{2x}+1)  (correct limits for e^{2x} -> 0/inf)
__device__ __forceinline__ float ftanh(float x) {
  const float e2 = __expf(2.0f * x);
  return 1.0f - 2.0f * __builtin_amdgcn_rcpf(e2 + 1.0f);
}

// ---------------------------------------------------------------------------
// Kernel 1: gold path score, one block (256 threads) per dialog.
// ---------------------------------------------------------------------------
__global__ void crf_gold_kernel(const float* __restrict__ trans,
                                const float* __restrict__ w,
                                const float* __restrict__ mult,
                                const float* __restrict__ out_table,
                                const float* __restrict__ bias,
                                const int*   __restrict__ sentence,
                                const int*   __restrict__ tags,
                                float*       __restrict__ gold) {
  const int b   = blockIdx.x;
  const int tid = threadIdx.x;
  __shared__ float red[256];

  float denom[4];
#pragma unroll
  for (int i = 0; i < 4; ++i)
    denom[i] = __expf(mult[0 * 4 + i]) + __expf(mult[1 * 4 + i]) +
               __expf(mult[2 * 4 + i]) + __expf(mult[3 * 4 + i]);

  float s = 0.0f;
  for (int t = tid; t < TLEN; t += 256) {
    const int tg = tags[b * TLEN + t];
    const int pv = (t == 0) ? STARTS : tags[b * TLEN + t - 1];
    float tr = trans[tg * NTAG + pv];
    if (pv < 4) {  // tags always < 4; bias term active for t > 0
      const float bt = bias[b * TLEN + t];
      const float Mv = (tg == pv) ? -1.0f
                                  : (__expf(mult[tg * 4 + pv]) *
                                     __builtin_amdgcn_rcpf(denom[pv]));
      tr += bt * ftanh((bt - 0.5f) * w[pv]) * Mv;
    }
    s += tr + out_table[sentence[b * TLEN + t] * 4 + tg];
  }
  red[tid] = s;
  __syncthreads();
  for (int off = 128; off > 0; off >>= 1) {
    if (tid < off) red[tid] += red[tid + off];
    __syncthreads();
  }
  if (tid == 0)
    gold[b] = red[0] + trans[STOPS * NTAG + tags[b * TLEN + TLEN - 1]];
}

// ---------------------------------------------------------------------------
// Kernel 2: per-(dialog, chunk) transfer matrix via WMMA chain in scaled
// linear space. One wave per chunk; 512*32 = 16384 waves.
//   P <- E_t x P (16x16 tile, 6x6 live); E built directly in A layout;
//   D->B relayout via SWAPX16 ds_swizzle (partner lane is lane^16).
// ---------------------------------------------------------------------------
__global__ __launch_bounds__(32)
void crf_chunk_kernel(const float* __restrict__ trans,
                      const float* __restrict__ w,
                      const float* __restrict__ mult,
                      const float* __restrict__ out_table,
                      const float* __restrict__ bias,
                      const int*   __restrict__ sentence,
                      float*       __restrict__ logG) {
  const int wid  = blockIdx.x;
  const int b    = wid / NCH;
  const int c    = wid % NCH;
  const int lane = threadIdx.x;
  const int col  = lane & 15;   // A-row (j) and B/D-column index
  const int h    = lane >> 4;   // lane half selects K pair

  // Uniform scale bound: max(transitions) + 1 (|M*shift|<=1) + 8 (feat bound)
  float mt = -1e30f;
  for (int k = 0; k < 36; ++k) mt = fmaxf(mt, trans[k]);
  const float SMAX = mt + 9.0f;

  // Chunk-invariant per-lane A-operand parameters; fold exp(T - SMAX) now.
  // A1: E columns i = 2h + r (0..3);  A2: E columns i = 4 + 2h + r (4,5 @ h=0)
  float E1b[2], Mw1[2], Wv1[2], E2b[2];
#pragma unroll
  for (int r = 0; r < 2; ++r) {
    const int i = 2 * h + r;                       // 0..3
    const int j = col;
    E1b[r] = (j < 6) ? __expf(trans[j * NTAG + i] - SMAX) : 0.0f;
    Wv1[r] = w[i];
    if (j < 4) {
      const float denom = __expf(mult[0 * 4 + i]) + __expf(mult[1 * 4 + i]) +
                          __expf(mult[2 * 4 + i]) + __expf(mult[3 * 4 + i]);
      Mw1[r] = (j == i) ? -1.0f
                        : (__expf(mult[j * 4 + i]) * __builtin_amdgcn_rcpf(denom));
    } else {
      Mw1[r] = 0.0f;
    }
    const int i2 = 4 + 2 * h + r;                  // 4,5 (h=0) else dead
    E2b[r] = (j < 6 && i2 < 6) ? __expf(trans[j * NTAG + i2] - SMAX) : 0.0f;
  }

  // P = identity on the live 6x6 block (D layout: lane holds P[r+8h, col])
  v8f acc = {0.f, 0.f, 0.f, 0.f, 0.f, 0.f, 0.f, 0.f};
#pragma unroll
  for (int r = 0; r < 8; ++r)
    if ((r + 8 * h) == col && col < 6) acc[r] = 1.0f;

  float L = 0.0f;
  const int t0 = c * CHUNK;

  for (int ms = 0; ms < CHUNK / 4; ++ms) {
    // ---- cluster the global loads for 4 steps up front ----
    float btv[4], fjv[4];
    int   sntv[4];
#pragma unroll
    for (int k = 0; k < 4; ++k) {
      const int t = t0 + 4 * ms + k;
      btv[k]  = bias[b * TLEN + t];
      sntv[k] = sentence[b * TLEN + t];
    }
#pragma unroll
    for (int k = 0; k < 4; ++k) {
      const int t = t0 + 4 * ms + k;
      if (col < 4)       fjv[k] = out_table[sntv[k] * 4 + col];
      else if (col == 4) fjv[k] = (t == 0)        ? 3.0f : -3.0f;
      else if (col == 5) fjv[k] = (t == TLEN - 1) ? 3.0f : -3.0f;
      else               fjv[k] = 0.0f;
    }

    // ---- 4 chained steps ----
#pragma unroll
    for (int k = 0; k < 4; ++k) {
      const float bt = btv[k];
      const float ef = __expf(fjv[k]);

      v2f a1, a2;
#pragma unroll
      for (int r = 0; r < 2; ++r) {
        const float sh = bt * ftanh((bt - 0.5f) * Wv1[r]);
        a1[r] = E1b[r] * ef * __expf(Mw1[r] * sh);
        a2[r] = E2b[r] * ef;
      }

      // D->B relayout: upper half reads partner (lane^16), lower half own regs
      v2f b1, b2;
#pragma unroll
      for (int r = 0; r < 2; ++r) {
        const float p1 = swz<SWZ_X16>(acc[2 + r]);
        const float p2 = swz<SWZ_X16>(acc[6 + r]);
        b1[r] = h ? p1 : acc[r];
        b2[r] = h ? p2 : acc[4 + r];
      }

      v8f z = {0.f, 0.f, 0.f, 0.f, 0.f, 0.f, 0.f, 0.f};
      v8f d = __builtin_amdgcn_wmma_f32_16x16x4_f32(false, a1, false, b1,
                                                    (short)0, z, false, false);
      d = __builtin_amdgcn_wmma_f32_16x16x4_f32(false, a2, false, b2,
                                                (short)0, d, false, false);
      acc = d;
    }

    // ---- renormalize (uniform branch point, EXEC all-ones throughout) ----
    float m = 0.0f;
#pragma unroll
    for (int k = 0; k < 8; ++k) m = fmaxf(m, acc[k]);
    // live block is lanes 0..15; butterfly within 16-lane halves
    m = fmaxf(m, swz<SWZ_X8>(m));
    m = fmaxf(m, swz<SWZ_X4>(m));
    m = fmaxf(m, swz<SWZ_X2>(m));
    m = fmaxf(m, swz<SWZ_X1>(m));
    m = fmaxf(m, 1e-30f);
    const float inv = __builtin_amdgcn_rcpf(m);
#pragma unroll
    for (int k = 0; k < 8; ++k) acc[k] *= inv;
    L += __logf(m);
  }
  L += (float)CHUNK * SMAX;

  // Live 6x6 block sits in h==0 lanes 0..5, regs 0..5: P[r, col]
  if (h == 0 && col < 6) {
    float* G = logG + (size_t)(b * NCH + c) * 36;
#pragma unroll
    for (int r = 0; r < 6; ++r)
      G[r * 6 + col] = fmaxf(__logf(acc[r]), -1e30f) + L;
  }
}

// ---------------------------------------------------------------------------
// Kernel 3: fold 32 chunk matrices per dialog in the log semiring (one lane
// per dialog), finish with STOP row, subtract gold.
// ---------------------------------------------------------------------------
__global__ void crf_combine_kernel(const float* __restrict__ trans,
                                   const float* __restrict__ logG,
                                   const float* __restrict__ gold,
                                   float* __restrict__ out) {
  const int b = blockIdx.x * blockDim.x + threadIdx.x;
  if (b >= BSZ) return;

  float alpha[NTAG];
#pragma unroll
  for (int j = 0; j < NTAG; ++j) alpha[j] = (j == STARTS) ? 0.0f : NEGV;

  for (int c = 0; c < NCH; ++c) {
    const float* G = logG + (size_t)(b * NCH + c) * 36;
    float na[NTAG];
#pragma unroll
    for (int j = 0; j < NTAG; ++j) {
      float m = -1e30f;
#pragma unroll
      for (int i = 0; i < NTAG; ++i) m = fmaxf(m, G[j * 6 + i] + alpha[i]);
      if (m > -1e28f) {
        float ss = 0.0f;
#pragma unroll
        for (int i = 0; i < NTAG; ++i) ss += __expf(G[j * 6 + i] + alpha[i] - m);
        na[j] = m + __logf(ss);
      } else {
        na[j] = -1e30f;
      }
    }
#pragma unroll
    for (int j = 0; j < NTAG; ++j) alpha[j] = na[j];
  }

  float m = -1e30f;
#pragma unroll
  for (int j = 0; j < NTAG; ++j)
    m = fmaxf(m, alpha[j] + trans[STOPS * NTAG + j]);
  float ss = 0.0f;
#pragma unroll
  for (int j = 0; j < NTAG; ++j)
    ss += __expf(alpha[j] + trans[STOPS * NTAG + j] - m);
  out[b] = (m + __logf(ss)) - gold[b];
}

// ---------------------------------------------------------------------------
extern "C" void kernel_launch(void* const* d_in, const int* in_sizes, int n_in,
                              void* d_out, int out_size, void* d_ws, size_t ws_size,
                              hipStream_t stream) {
  const float* trans     = (const float*)d_in[0];  // [6,6]
  const float* w         = (const float*)d_in[1];  // [4]
  const float* mult      = (const float*)d_in[2];  // [4,4]
  const float* out_table = (const float*)d_in[3];  // [V,4]
  const float* bias      = (const float*)d_in[4];  // [B,T]
  const int*   sentence  = (const int*)d_in[5];    // [B,T]
  const int*   tags      = (const int*)d_in[6];    // [B,T]
  float* out = (float*)d_out;                      // [B]

  float* gold = (float*)d_ws;                      // [B]
  float* logG = gold + BSZ;                        // [B, NCH, 6, 6]

  crf_gold_kernel<<<BSZ, 256, 0, stream>>>(trans, w, mult, out_table, bias,
                                           sentence, tags, gold);
  crf_chunk_kernel<<<BSZ * NCH, 32, 0, stream>>>(trans, w, mult, out_table,
                                                 bias, sentence, logG);
  crf_combine_kernel<<<(BSZ + 127) / 128, 128, 0, stream>>>(trans, logG, gold, out);
}